// TripletAttnDynMessage_77979426226458
// MI455X (gfx1250) — compile-verified
//
#include <hip/hip_runtime.h>
#include <math.h>

#define NN 256      // nodes
#define EE 320      // edges
#define DE 640      // 2E directed edges
#define HV 128      // hv/he/mv/me dim
#define KA 264      // 2*HV + 8
#define KB 136      // HV + 8
#define DATT 416

typedef __attribute__((ext_vector_type(2))) float v2f;
typedef __attribute__((ext_vector_type(8))) float v8f;

// ---------------------------------------------------------------- decode ----
// Directed edge k: k<E uses (vew1=u, vew2=v); k>=E swaps. Recover indices,
// compute normalized displacement and dis_ftr = tanh(tanh(dis)*Wdis + bdis).
__global__ void k_decode(const float* __restrict__ vew1, const float* __restrict__ vew2,
                         const float* __restrict__ qf, const float* __restrict__ Wdis,
                         const float* __restrict__ bdis,
                         int* __restrict__ iU, int* __restrict__ iV,
                         float* __restrict__ nd, float* __restrict__ dft) {
  int k = blockIdx.x;
  int e = (k < EE) ? k : k - EE;
  const float* mu = (k < EE) ? vew1 : vew2;
  const float* mv = (k < EE) ? vew2 : vew1;
  __shared__ int su, sv;
  if (threadIdx.x == 0) { su = 0; sv = 0; }
  __syncthreads();
  for (int n = (int)threadIdx.x; n < NN; n += 32) {
    if (mu[n * EE + e] > 0.5f) su = n;
    if (mv[n * EE + e] > 0.5f) sv = n;
  }
  __syncthreads();
  if (threadIdx.x == 0) {
    int U = su, V = sv;
    iU[k] = U; iV[k] = V;
    float dx = qf[V * 3 + 0] - qf[U * 3 + 0];
    float dy = qf[V * 3 + 1] - qf[U * 3 + 1];
    float dz = qf[V * 3 + 2] - qf[U * 3 + 2];
    float dis = sqrtf(dx * dx + dy * dy + dz * dz) + 1e-6f;
    nd[k * 3 + 0] = dx / dis;
    nd[k * 3 + 1] = dy / dis;
    nd[k * 3 + 2] = dz / dis;
    float td = tanhf(dis);
    for (int j = 0; j < 8; ++j) dft[k * 8 + j] = tanhf(td * Wdis[j] + bdis[j]);
  }
}

// -------------------------------------------------------------- align_e -----
__global__ void k_alignE(const float* __restrict__ he, const float* __restrict__ Wal,
                         const float* __restrict__ bal, float* __restrict__ alE) {
  int e = blockIdx.x, t = threadIdx.x;
  __shared__ float red[128];
  red[t] = he[e * HV + t] * Wal[t];
  __syncthreads();
  for (int s = 64; s > 0; s >>= 1) { if (t < s) red[t] += red[t + s]; __syncthreads(); }
  if (t == 0) alE[e] = red[0] + bal[0];
}

// ---------------------------------------------------- packed GEMM inputs ----
// Xa[k] = [hv[V] | dis_ftr | hv[U]]  (264)   (== [vd_ftr, hv_u] of reference)
// Xm[k] = [hv[U] | dis_ftr | hv[V]]  (264)   (W_link input order)
__global__ void k_buildX(const float* __restrict__ hv, const int* __restrict__ iU,
                         const int* __restrict__ iV, const float* __restrict__ dft,
                         float* __restrict__ Xa, float* __restrict__ Xm) {
  int k = blockIdx.x;
  int U = iU[k], V = iV[k];
  for (int c = (int)threadIdx.x; c < KA; c += blockDim.x) {
    float av, am;
    if (c < HV)            { av = hv[V * HV + c];            am = hv[U * HV + c]; }
    else if (c < HV + 8)   { float d = dft[k * 8 + (c - HV)]; av = d; am = d; }
    else                   { av = hv[U * HV + (c - HV - 8)]; am = hv[V * HV + (c - HV - 8)]; }
    Xa[k * KA + c] = av;
    Xm[k * KA + c] = am;
  }
}

// --------------------------------------------------- WMMA f32 GEMMs ---------
// C[M,N] = A[M,K] @ B^T, B given row-major [N,K] (weight layout), one wave
// per 16x16 tile. A frag: lane l -> row l&15, regs hold K = 2*(l>>4)+{0,1}.
// B frag (4x16 KxN): lane l -> col l&15, regs hold K = 2*(l>>4)+{0,1}.
// C/D frag: lane l reg r -> row r+8*(l>>4), col l&15.
__global__ void k_gemm_wt(const float* __restrict__ A, int lda,
                          const float* __restrict__ B, int ldb,
                          float* __restrict__ C, int ldc, int K) {
  int m0 = blockIdx.x * 16, n0 = blockIdx.y * 16;
  int l = (int)threadIdx.x;
  int rc = l & 15;
  int kh = (l >> 4) << 1;                   // 0 or 2
  const float* ap = A + (size_t)(m0 + rc) * lda + kh;
  const float* bp = B + (size_t)(n0 + rc) * ldb + kh;
  v8f acc = {};
  for (int k0 = 0; k0 < K; k0 += 4) {
    v2f a; a.x = ap[k0]; a.y = ap[k0 + 1];
    v2f b; b.x = bp[k0]; b.y = bp[k0 + 1];
    acc = __builtin_amdgcn_wmma_f32_16x16x4_f32(false, a, false, b, (short)0, acc,
                                                false, false);
  }
  int rh = l >> 4;
  for (int r = 0; r < 8; ++r)
    C[(size_t)(m0 + r + 8 * rh) * ldc + n0 + rc] = acc[r];
}

// C[M,N] = A[M,K] @ B[K,N], B row-major [K,N].
__global__ void k_gemm_nn(const float* __restrict__ A, int lda,
                          const float* __restrict__ B, int ldb,
                          float* __restrict__ C, int ldc, int K) {
  int m0 = blockIdx.x * 16, n0 = blockIdx.y * 16;
  int l = (int)threadIdx.x;
  int rc = l & 15;
  int kh = (l >> 4) << 1;
  const float* ap = A + (size_t)(m0 + rc) * lda + kh;
  v8f acc = {};
  for (int k0 = 0; k0 < K; k0 += 4) {
    v2f a; a.x = ap[k0]; a.y = ap[k0 + 1];
    v2f b;
    b.x = B[(size_t)(k0 + kh) * ldb + n0 + rc];
    b.y = B[(size_t)(k0 + kh + 1) * ldb + n0 + rc];
    acc = __builtin_amdgcn_wmma_f32_16x16x4_f32(false, a, false, b, (short)0, acc,
                                                false, false);
  }
  int rh = l >> 4;
  for (int r = 0; r < 8; ++r)
    C[(size_t)(m0 + r + 8 * rh) * ldc + n0 + rc] = acc[r];
}

// -------------------------------------------------------------- attend ------
// attend[i,m] = max(0, max_{j: U(j)==U(i)} relu(a_i[i,m]+b_j[j,m]+f_m(ang)))
// f_m(x) = sum_k tanh(x*Wang[k]+bang[k]) * W_att[m, 400+k]
__global__ void k_attend(const int* __restrict__ iU, const float* __restrict__ nd,
                         const float* __restrict__ Ai, const float* __restrict__ Bj,
                         const float* __restrict__ Watt, const float* __restrict__ Wang,
                         const float* __restrict__ bang, float* __restrict__ Att) {
  int i = blockIdx.x, t = threadIdx.x;
  __shared__ int cnt;
  __shared__ int list[DE];
  __shared__ float tk[16];
  __shared__ float ndi[3];
  if (t == 0) cnt = 0;
  if (t < 3) ndi[t] = nd[i * 3 + t];
  __syncthreads();
  int Ui = iU[i];
  for (int j = t; j < DE; j += 128)
    if (iU[j] == Ui) { int p = atomicAdd(&cnt, 1); list[p] = j; }
  __syncthreads();
  float w3[16];
  for (int k = 0; k < 16; ++k) w3[k] = Watt[t * DATT + 400 + k];
  float aim = Ai[i * HV + t];
  float acc = 0.0f;
  int L = cnt;
  for (int p = 0; p < L; ++p) {
    int j = list[p];
    if (t < 16) {
      float ang = ndi[0] * nd[j * 3 + 0] + ndi[1] * nd[j * 3 + 1] + ndi[2] * nd[j * 3 + 2];
      tk[t] = tanhf(ang * Wang[t] + bang[t]);
    }
    __syncthreads();
    float c = 0.0f;
    for (int k = 0; k < 16; ++k) c += tk[k] * w3[k];
    float s = aim + Bj[j * HV + t] + c;
    acc = fmaxf(acc, fmaxf(s, 0.0f));
    __syncthreads();
  }
  Att[i * HV + t] = acc;
}

// ---------------------------------------------------- dense align softmax ---
// logits[n,k] = (V(k)==n) ? align_e[k%E] : -1e6 ; row softmax (matches jax:
// exp underflow -> 0; empty row -> uniform 1/640).
__global__ void k_softmax(const int* __restrict__ iV, const float* __restrict__ alE,
                          float* __restrict__ AlD) {
  int n = blockIdx.x, t = threadIdx.x;
  __shared__ float red[256];
  float mx = -3.0e38f;
  for (int k = t; k < DE; k += 256) {
    float lg = (iV[k] == n) ? alE[k < EE ? k : k - EE] : -1.0e6f;
    mx = fmaxf(mx, lg);
  }
  red[t] = mx; __syncthreads();
  for (int s = 128; s > 0; s >>= 1) { if (t < s) red[t] = fmaxf(red[t], red[t + s]); __syncthreads(); }
  float M = red[0]; __syncthreads();
  float sm = 0.0f;
  for (int k = t; k < DE; k += 256) {
    float lg = (iV[k] == n) ? alE[k < EE ? k : k - EE] : -1.0e6f;
    sm += expf(lg - M);
  }
  red[t] = sm; __syncthreads();
  for (int s = 128; s > 0; s >>= 1) { if (t < s) red[t] += red[t + s]; __syncthreads(); }
  float S = red[0];
  for (int k = t; k < DE; k += 256) {
    float lg = (iV[k] == n) ? alE[k < EE ? k : k - EE] : -1.0e6f;
    AlD[n * DE + k] = expf(lg - M) / S;
  }
}

// ------------------------------------------------------------- epilogues ----
__global__ void k_mv_out(const float* __restrict__ Mvr, float* __restrict__ out) {
  int idx = blockIdx.x * blockDim.x + threadIdx.x;  // 256*128
  float x = Mvr[idx];
  out[idx] = (x > 0.0f) ? x : (expf(x) - 1.0f);     // elu
}

__global__ void k_me_out(const float* __restrict__ Me2, const float* __restrict__ blink,
                         float* __restrict__ out) {
  int idx = blockIdx.x * blockDim.x + threadIdx.x;  // 320*128
  int m = idx & (HV - 1);
  float s = Me2[idx] + Me2[idx + EE * HV] + 2.0f * blink[m];
  out[idx] = (s > 0.0f) ? s : 0.01f * s;            // leaky_relu
}

// ---------------------------------------------------------------- launch ----
extern "C" void kernel_launch(void* const* d_in, const int* in_sizes, int n_in,
                              void* d_out, int out_size, void* d_ws, size_t ws_size,
                              hipStream_t stream) {
  const float* hv   = (const float*)d_in[0];
  const float* he   = (const float*)d_in[1];
  // d_in[2] = p_ftr (unused by reference)
  const float* qf   = (const float*)d_in[3];
  const float* vew1 = (const float*)d_in[4];
  const float* vew2 = (const float*)d_in[5];
  // d_in[6], d_in[7] = veb1/veb2 (reconstructed exactly as (vew-1)*1e6 masks)
  const float* Wdis = (const float*)d_in[8];
  const float* bdis = (const float*)d_in[9];
  const float* Wang = (const float*)d_in[10];
  const float* bang = (const float*)d_in[11];
  const float* Watt = (const float*)d_in[12];   // [128, 416]
  const float* Wal  = (const float*)d_in[13];   // [1, 128]
  const float* bal  = (const float*)d_in[14];
  const float* Wlnk = (const float*)d_in[15];   // [128, 264]
  const float* blnk = (const float*)d_in[16];

  char* w = (char*)d_ws;
  int*   iU  = (int*)w;    w += DE * sizeof(int);
  int*   iV  = (int*)w;    w += DE * sizeof(int);
  float* nd  = (float*)w;  w += DE * 3 * sizeof(float);
  float* dft = (float*)w;  w += DE * 8 * sizeof(float);
  float* alE = (float*)w;  w += EE * sizeof(float);
  float* Xa  = (float*)w;  w += (size_t)DE * KA * sizeof(float);
  float* Xm  = (float*)w;  w += (size_t)DE * KA * sizeof(float);
  float* Ai  = (float*)w;  w += (size_t)DE * HV * sizeof(float);
  float* Bj  = (float*)w;  w += (size_t)DE * HV * sizeof(float);
  float* Me2 = (float*)w;  w += (size_t)DE * HV * sizeof(float);
  float* Att = (float*)w;  w += (size_t)DE * HV * sizeof(float);
  float* AlD = (float*)w;  w += (size_t)NN * DE * sizeof(float);
  float* Mvr = (float*)w;  w += (size_t)NN * HV * sizeof(float);

  float* mv_out = (float*)d_out;            // [256,128]
  float* me_out = (float*)d_out + NN * HV;  // [320,128]

  k_decode<<<DE, 32, 0, stream>>>(vew1, vew2, qf, Wdis, bdis, iU, iV, nd, dft);
  k_alignE<<<EE, 128, 0, stream>>>(he, Wal, bal, alE);
  k_buildX<<<DE, 256, 0, stream>>>(hv, iU, iV, dft, Xa, Xm);

  // a_i = Xa @ W_att[:, :264]^T        (640x264x128)
  k_gemm_wt<<<dim3(DE / 16, HV / 16), 32, 0, stream>>>(Xa, KA, Watt, DATT, Ai, HV, KA);
  // b_j = Xa[:, :136] @ W_att[:, 264:400]^T
  k_gemm_wt<<<dim3(DE / 16, HV / 16), 32, 0, stream>>>(Xa, KA, Watt + KA, DATT, Bj, HV, KB);
  // me2raw = Xm @ W_link^T
  k_gemm_wt<<<dim3(DE / 16, HV / 16), 32, 0, stream>>>(Xm, KA, Wlnk, KA, Me2, HV, KA);

  k_attend<<<DE, 128, 0, stream>>>(iU, nd, Ai, Bj, Watt, Wang, bang, Att);
  k_softmax<<<NN, 256, 0, stream>>>(iV, alE, AlD);

  // mv_raw = align_dense @ attend       (256x640x128, B row-major [K,N])
  k_gemm_nn<<<dim3(NN / 16, HV / 16), 32, 0, stream>>>(AlD, DE, Att, HV, Mvr, HV, DE);

  k_mv_out<<<(NN * HV) / 256, 256, 0, stream>>>(Mvr, mv_out);
  k_me_out<<<(EE * HV) / 256, 256, 0, stream>>>(Me2, blnk, me_out);
}